// Net_16673063043120
// MI455X (gfx1250) — compile-verified
//
#include <hip/hip_runtime.h>
#include <hip/hip_bf16.h>
#include <math.h>

#define N_NODES  50000
#define E_EDGES  800000
#define E_TOT    (E_EDGES + N_NODES)   // 850000 with self-loops
#define F_IN     256
#define HID      32
#define HEADS    8
#define F1       (HEADS * HID)         // 256
#define CLASSES  40
#define NEG_SLOPE 0.2f

typedef __attribute__((ext_vector_type(2))) float v2f;
typedef __attribute__((ext_vector_type(8))) float v8f;

// ---------------------------------------------------------------------------
// Sign-aware float atomic max (correct for mixed signs with -inf init)
// ---------------------------------------------------------------------------
__device__ __forceinline__ float atomicMaxF(float* addr, float value) {
    if (value >= 0.0f)
        return __int_as_float(atomicMax((int*)addr, __float_as_int(value)));
    else
        return __uint_as_float(atomicMin((unsigned int*)addr, __float_as_uint(value)));
}

__global__ void fill_f32(float* __restrict__ p, float v, int n) {
    int i = blockIdx.x * blockDim.x + threadIdx.x;
    if (i < n) p[i] = v;
}

// ---------------------------------------------------------------------------
// FP32 WMMA GEMM:  C[M,Ncols] = A[M,K] * B[K,Ncols]   (row-major)
// One wave32 computes a 16x64 strip: 4 n-tiles, 4 accumulators, so each
// b64 A-fragment load feeds 4 V_WMMA_F32_16X16X4_F32 ops.
//
// Ragged Ncols: out-of-range lanes clamp their B column pointer to a valid
// in-bounds column (Ncols-1). In a matmul, B column n only contributes to
// output column n, and those lanes' stores are masked — so the garbage never
// reaches memory and the hot loop stays branch-free with full EXEC.
//
// Fragments (16x16x4 f32, wave32):
//   A (16x4): lane l -> m = l&15, kb = (l>>4)*2, VGPRs {A[m,kb], A[m,kb+1]}
//   B (4x16): lane l -> n = l&15, kb = (l>>4)*2, VGPRs {B[kb,n], B[kb+1,n]}
//   C/D     : VGPR v -> M = v + (l>>4)*8, N = l&15
// ---------------------------------------------------------------------------
__global__ void gemm_wmma_f32(const float* __restrict__ A,
                              const float* __restrict__ B,
                              float* __restrict__ C,
                              int M, int K, int Ncols, int ngroups) {
    const int lane  = threadIdx.x & 31;
    const int wave  = threadIdx.x >> 5;
    const int tile  = blockIdx.x * (blockDim.x >> 5) + wave;
    const int total = (M >> 4) * ngroups;
    if (tile >= total) return;                 // wave-uniform exit

    const int mt = tile / ngroups;
    const int g  = tile % ngroups;

    const int l15 = lane & 15;
    const int kb  = (lane >> 4) << 1;          // 0 or 2
    const int mrow = (mt << 4) + l15;
    const float* __restrict__ Arow = A + (size_t)mrow * K + kb;

    int  ncol[4];
    bool nOK[4];
    const float* Bcol[4];
#pragma unroll
    for (int t = 0; t < 4; ++t) {
        ncol[t] = (g << 6) + (t << 4) + l15;
        nOK[t]  = (ncol[t] < Ncols);
        Bcol[t] = B + (size_t)kb * Ncols + (nOK[t] ? ncol[t] : (Ncols - 1));
    }

    v8f acc[4] = {v8f{}, v8f{}, v8f{}, v8f{}};

    for (int k0 = 0; k0 < K; k0 += 4) {
        v2f a;
        a.x = Arow[k0];                        // contiguous pair -> b64 load
        a.y = Arow[k0 + 1];
        const size_t r0 = (size_t)k0 * Ncols;
        const size_t r1 = r0 + Ncols;
        v2f b[4];
#pragma unroll
        for (int t = 0; t < 4; ++t) {
            b[t].x = Bcol[t][r0];
            b[t].y = Bcol[t][r1];
        }
#pragma unroll
        for (int t = 0; t < 4; ++t)
            acc[t] = __builtin_amdgcn_wmma_f32_16x16x4_f32(
                         false, a, false, b[t], (short)0, acc[t], false, false);
    }

    const int half = lane >> 4;                // 0 or 1
#pragma unroll
    for (int t = 0; t < 4; ++t) {
        if (nOK[t]) {
#pragma unroll
            for (int v = 0; v < 8; ++v) {
                const int row = (mt << 4) + v + half * 8;
                C[(size_t)row * Ncols + ncol[t]] = acc[t][v];
            }
        }
    }
}

// ---------------------------------------------------------------------------
// alpha_src[n,h] = sum_c feat[n,h,c]*a_src[h,c]   (and same for a_dst)
// ---------------------------------------------------------------------------
__global__ void alpha_k(const float* __restrict__ feat,
                        const float* __restrict__ a_src,
                        const float* __restrict__ a_dst,
                        float* __restrict__ asrc, float* __restrict__ adst,
                        int heads, int ch) {
    int i = blockIdx.x * blockDim.x + threadIdx.x;
    if (i >= N_NODES * heads) return;
    const int n = i / heads, hd = i % heads;
    const float* hp = feat + (size_t)n * heads * ch + hd * ch;
    const float* as = a_src + hd * ch;
    const float* ad = a_dst + hd * ch;
    float s1 = 0.0f, s2 = 0.0f;
    for (int c = 0; c < ch; ++c) { s1 += hp[c] * as[c]; s2 += hp[c] * ad[c]; }
    asrc[i] = s1; adst[i] = s2;
}

__device__ __forceinline__ void edge_ids(const int* __restrict__ ei, int e,
                                         int& s, int& d) {
    if (e < E_EDGES) { s = ei[e]; d = ei[E_EDGES + e]; }
    else             { s = e - E_EDGES; d = s; }       // appended self-loops
}

// Pass A: segment max of leaky_relu logits over destination nodes
__global__ void edge_max_k(const int* __restrict__ ei,
                           const float* __restrict__ asrc,
                           const float* __restrict__ adst,
                           float* __restrict__ emax, int heads) {
    int idx = blockIdx.x * blockDim.x + threadIdx.x;
    if (idx >= E_TOT * heads) return;
    const int e = idx / heads, hd = idx % heads;
    int s, d; edge_ids(ei, e, s, d);
    float v = asrc[s * heads + hd] + adst[d * heads + hd];
    v = (v > 0.0f) ? v : NEG_SLOPE * v;
    atomicMaxF(&emax[d * heads + hd], v);
}

// Pass B: denom += exp(e - emax);  out[d] += exp(e - emax) * feat[s]
// (normalization by denom is deferred to a per-node pass — algebraically
//  identical to the reference's alpha = ex/denom followed by segment_sum)
__global__ void edge_accum_k(const int* __restrict__ ei,
                             const float* __restrict__ asrc,
                             const float* __restrict__ adst,
                             const float* __restrict__ emax,
                             float* __restrict__ denom,
                             const float* __restrict__ feat,
                             float* __restrict__ out,
                             int heads, int ch) {
    int idx = blockIdx.x * blockDim.x + threadIdx.x;
    if (idx >= E_TOT * heads) return;
    const int e = idx / heads, hd = idx % heads;
    int s, d; edge_ids(ei, e, s, d);
    float v = asrc[s * heads + hd] + adst[d * heads + hd];
    v = (v > 0.0f) ? v : NEG_SLOPE * v;
    float m = emax[d * heads + hd];
    if (!isfinite(m)) m = 0.0f;                 // reference's finite guard
    const float ex = expf(v - m);
    atomicAdd(&denom[d * heads + hd], ex);
    const float* fs = feat + (size_t)s * heads * ch + hd * ch;
    float*       od = out  + (size_t)d * heads * ch + hd * ch;
    for (int c = 0; c < ch; ++c) atomicAdd(&od[c], fs[c] * ex);
}

// Layer-1 epilogue: out /= (denom+1e-16); out += b1; ELU  (in place)
__global__ void norm_bias_elu_k(float* __restrict__ out1,
                                const float* __restrict__ denom,
                                const float* __restrict__ bias) {
    int i = blockIdx.x * blockDim.x + threadIdx.x;
    if (i >= N_NODES * F1) return;
    const int n = i / F1, j = i % F1, hd = j / HID;
    float v = out1[i] / (denom[n * HEADS + hd] + 1e-16f) + bias[j];
    out1[i] = (v > 0.0f) ? v : (expf(v) - 1.0f);
}

// Layer-2 epilogue: normalize, +b2, stable log_softmax over 40 classes
__global__ void final_k(const float* __restrict__ out2,
                        const float* __restrict__ den2,
                        const float* __restrict__ b2,
                        float* __restrict__ y) {
    int n = blockIdx.x * blockDim.x + threadIdx.x;
    if (n >= N_NODES) return;
    float o[CLASSES];
    const float inv = 1.0f / (den2[n] + 1e-16f);
    float mx = -INFINITY;
    for (int c = 0; c < CLASSES; ++c) {
        float v = out2[(size_t)n * CLASSES + c] * inv + b2[c];
        o[c] = v; mx = fmaxf(mx, v);
    }
    float ss = 0.0f;
    for (int c = 0; c < CLASSES; ++c) ss += expf(o[c] - mx);
    const float lse = logf(ss);
    for (int c = 0; c < CLASSES; ++c)
        y[(size_t)n * CLASSES + c] = o[c] - mx - lse;
}

static inline int cdiv(long long a, long long b) { return (int)((a + b - 1) / b); }

extern "C" void kernel_launch(void* const* d_in, const int* in_sizes, int n_in,
                              void* d_out, int out_size, void* d_ws, size_t ws_size,
                              hipStream_t stream) {
    (void)in_sizes; (void)n_in; (void)out_size; (void)ws_size;
    const float* x      = (const float*)d_in[0];
    const int*   ei     = (const int*)  d_in[1];
    const float* W1     = (const float*)d_in[2];
    const float* a_src1 = (const float*)d_in[3];
    const float* a_dst1 = (const float*)d_in[4];
    const float* b1     = (const float*)d_in[5];
    const float* W2     = (const float*)d_in[6];
    const float* a_src2 = (const float*)d_in[7];
    const float* a_dst2 = (const float*)d_in[8];
    const float* b2     = (const float*)d_in[9];
    float* y = (float*)d_out;

    // workspace carve-up (floats)
    float* ws    = (float*)d_ws;      size_t o = 0;
    float* h1    = ws + o; o += (size_t)N_NODES * F1;      // 12.8M
    float* out1  = ws + o; o += (size_t)N_NODES * F1;      // 12.8M (becomes h2)
    float* asrc1 = ws + o; o += (size_t)N_NODES * HEADS;
    float* adst1 = ws + o; o += (size_t)N_NODES * HEADS;
    float* emax1 = ws + o; o += (size_t)N_NODES * HEADS;
    float* den1  = ws + o; o += (size_t)N_NODES * HEADS;
    float* g2    = ws + o; o += (size_t)N_NODES * CLASSES; // 2M
    float* out2  = ws + o; o += (size_t)N_NODES * CLASSES; // 2M
    float* asrc2 = ws + o; o += (size_t)N_NODES;
    float* adst2 = ws + o; o += (size_t)N_NODES;
    float* emax2 = ws + o; o += (size_t)N_NODES;
    float* den2  = ws + o; o += (size_t)N_NODES;

    const int B = 256;                 // 8 wave32s per block

    // ---- init accumulators (harness does not re-zero between replays) ----
    fill_f32<<<cdiv((long long)N_NODES * F1, B), B, 0, stream>>>(out1, 0.0f, N_NODES * F1);
    fill_f32<<<cdiv(N_NODES * HEADS, B), B, 0, stream>>>(emax1, -INFINITY, N_NODES * HEADS);
    fill_f32<<<cdiv(N_NODES * HEADS, B), B, 0, stream>>>(den1, 0.0f, N_NODES * HEADS);
    fill_f32<<<cdiv((long long)N_NODES * CLASSES, B), B, 0, stream>>>(out2, 0.0f, N_NODES * CLASSES);
    fill_f32<<<cdiv(N_NODES, B), B, 0, stream>>>(emax2, -INFINITY, N_NODES);
    fill_f32<<<cdiv(N_NODES, B), B, 0, stream>>>(den2, 0.0f, N_NODES);

    // ---- layer 1 ----
    {   // h1 = x @ W1  (M=50000, K=256, N=256 -> 4 column groups of 64)
        const int ngroups = F1 / 64;                        // 4
        const int tiles   = (N_NODES / 16) * ngroups;       // 12500 waves
        gemm_wmma_f32<<<cdiv(tiles, B / 32), B, 0, stream>>>(x, W1, h1,
                                                             N_NODES, F_IN, F1, ngroups);
    }
    alpha_k<<<cdiv(N_NODES * HEADS, B), B, 0, stream>>>(h1, a_src1, a_dst1,
                                                        asrc1, adst1, HEADS, HID);
    edge_max_k<<<cdiv((long long)E_TOT * HEADS, B), B, 0, stream>>>(ei, asrc1, adst1,
                                                                    emax1, HEADS);
    edge_accum_k<<<cdiv((long long)E_TOT * HEADS, B), B, 0, stream>>>(ei, asrc1, adst1,
                                                                      emax1, den1, h1, out1,
                                                                      HEADS, HID);
    norm_bias_elu_k<<<cdiv((long long)N_NODES * F1, B), B, 0, stream>>>(out1, den1, b1);

    // ---- layer 2 (out1 now holds ELU'd hidden features) ----
    {   // g2 = out1 @ W2  (Ncols=40 -> one 64-col group, lanes 40..63 masked)
        const int ngroups = 1;
        const int tiles   = (N_NODES / 16) * ngroups;       // 3125 waves
        gemm_wmma_f32<<<cdiv(tiles, B / 32), B, 0, stream>>>(out1, W2, g2,
                                                             N_NODES, F1, CLASSES, ngroups);
    }
    alpha_k<<<cdiv(N_NODES, B), B, 0, stream>>>(g2, a_src2, a_dst2,
                                                asrc2, adst2, 1, CLASSES);
    edge_max_k<<<cdiv(E_TOT, B), B, 0, stream>>>(ei, asrc2, adst2, emax2, 1);
    edge_accum_k<<<cdiv(E_TOT, B), B, 0, stream>>>(ei, asrc2, adst2, emax2, den2,
                                                   g2, out2, 1, CLASSES);
    final_k<<<cdiv(N_NODES, B), B, 0, stream>>>(out2, den2, b2, y);
}